// LabAlignmentModule_48120813585001
// MI455X (gfx1250) — compile-verified
//
#include <hip/hip_runtime.h>
#include <hip/hip_bf16.h>

typedef float v2f __attribute__((ext_vector_type(2)));
typedef float v4f __attribute__((ext_vector_type(4)));
typedef float v8f __attribute__((ext_vector_type(8)));

namespace {
constexpr int NB     = 64;        // batch
constexpr int NC     = 2048;      // channels
constexpr int HW     = 576;       // 24*24
constexpr int P4     = HW / 4;    // 144 float4 per spatial row
constexpr int CSPLIT = 16;        // channel-split for wave parallelism
constexpr int CCHUNK = NC / CSPLIT;  // 128
constexpr int LUME   = NB * HW;   // 36864 lum elements per tensor
}

// ---------------------------------------------------------------------------
// Pass 1: streaming squared-reduction over channels, C split 16 ways.
// Each thread owns one float4 column (b, p4) for one tensor and one c-chunk.
// Consecutive threads -> consecutive p4 -> 512B contiguous per warp per iter.
// Non-temporal loads: 604 MB stream, zero reuse, keep it out of L2.
// ---------------------------------------------------------------------------
__global__ void lum_partial_kernel(const float* __restrict__ fv,
                                   const float* __restrict__ fi,
                                   float* __restrict__ part) {
  const int q  = blockIdx.x * blockDim.x + threadIdx.x;  // 0..9215 (b,p4) column
  const int g  = blockIdx.y;                             // c-chunk
  const int t  = blockIdx.z;                             // tensor select
  const int b  = q / P4;
  const int p4 = q - b * P4;

  const v4f* __restrict__ src = (const v4f*)(t == 0 ? fv : fi);
  size_t base = (size_t)b * NC * P4 + (size_t)(g * CCHUNK) * P4 + p4;

  v4f acc = {0.f, 0.f, 0.f, 0.f};
#pragma unroll 8
  for (int c = 0; c < CCHUNK; ++c) {
    v4f v = __builtin_nontemporal_load(&src[base + (size_t)c * P4]);
    acc += v * v;
  }
  v4f* pp = (v4f*)part;
  pp[(size_t)(t * CSPLIT + g) * (LUME / 4) + q] = acc;
}

// ---------------------------------------------------------------------------
// Pass 2: fold the 16 channel-chunk partials -> lum[2][64][576].
// Fixed summation order => deterministic.
// ---------------------------------------------------------------------------
__global__ void lum_reduce_kernel(const float* __restrict__ part,
                                  float* __restrict__ lum) {
  const int idx = blockIdx.x * blockDim.x + threadIdx.x;
  if (idx >= 2 * LUME) return;
  const int t = idx / LUME;
  const int e = idx - t * LUME;
  float s = 0.f;
#pragma unroll
  for (int g = 0; g < CSPLIT; ++g)
    s += part[(size_t)(t * CSPLIT + g) * LUME + e];
  lum[idx] = s;
}

// ---------------------------------------------------------------------------
// Pass 3: one block, 16 waves. Wave w computes 16x16 tile (w/4, w%4) of
// cross = lum_v (64x576) * lum_i^T via v_wmma_f32_16x16x4_f32, K-loop = 144.
// Then masked MSE sum, reduced in fixed order (deterministic).
// ---------------------------------------------------------------------------
__global__ __launch_bounds__(512) void pair_loss_kernel(
    const float* __restrict__ lum, const int* __restrict__ labels,
    float* __restrict__ out) {
  __shared__ float mv2_s[NB];
  __shared__ float mi2_s[NB];
  __shared__ int   lab_s[NB];
  __shared__ int   np_s;
  __shared__ float lanesum[512];

  const int tid = threadIdx.x;
  if (tid < NB) lab_s[tid] = labels[tid];
  // row means of lum^2: mean_v2 / mean_i2
  if (tid < 2 * NB) {
    const float* row = lum + tid * HW;
    float s = 0.f;
    for (int p = 0; p < HW; ++p) { float x = row[p]; s += x * x; }
    s *= (1.0f / HW);
    if (tid < NB) mv2_s[tid] = s; else mi2_s[tid - NB] = s;
  }
  __syncthreads();
  if (tid == 0) {
    int np = 0;
    for (int i = 0; i < NB; ++i)
      for (int j = 0; j < NB; ++j)
        np += (i != j && lab_s[i] == lab_s[j]) ? 1 : 0;
    np_s = np;
  }
  __syncthreads();

  const int wave = tid >> 5, lane = tid & 31;
  const int it = wave >> 2, jt = wave & 3;     // tile coords in 4x4 grid
  const int half = lane >> 4, lm = lane & 15;

  // A fragment rows come from lum_v, B fragment from lum_i rows (B = lum_i^T).
  const float* __restrict__ arow = lum + (it * 16 + lm) * HW;
  const float* __restrict__ brow = lum + LUME + (jt * 16 + lm) * HW;

  v8f acc = {};
  for (int k0 = 0; k0 < HW; k0 += 4) {
    // 16x4 f32 A layout: lanes 0-15 hold K = k0,k0+1 ; lanes 16-31 hold k0+2,k0+3
    v2f a = *(const v2f*)(arow + k0 + 2 * half);
    v2f b = *(const v2f*)(brow + k0 + 2 * half);
    acc = __builtin_amdgcn_wmma_f32_16x16x4_f32(
        /*neg_a=*/false, a, /*neg_b=*/false, b,
        /*c_mod=*/(short)0, acc, /*reuse_a=*/false, /*reuse_b=*/false);
  }

  // Masked MSE sum for this lane's 8 C/D elements:
  // acc[r] = cross*HW at (M = it*16 + 8*half + r, N = jt*16 + lm)
  float s = 0.f;
  const float inv_hw = 1.0f / HW;
  const int j = jt * 16 + lm;
#pragma unroll
  for (int r = 0; r < 8; ++r) {
    const int i = it * 16 + half * 8 + r;
    if (i != j && lab_s[i] == lab_s[j])
      s += mv2_s[i] + mi2_s[j] - 2.0f * acc[r] * inv_hw;
  }
  lanesum[tid] = s;
  __syncthreads();
  if (tid == 0) {
    float total = 0.f;
    for (int k = 0; k < 512; ++k) total += lanesum[k];
    out[0] = (np_s > 0) ? (total / (float)np_s) : 0.0f;
  }
}

// ---------------------------------------------------------------------------
extern "C" void kernel_launch(void* const* d_in, const int* in_sizes, int n_in,
                              void* d_out, int out_size, void* d_ws, size_t ws_size,
                              hipStream_t stream) {
  const float* fv     = (const float*)d_in[0];
  const float* fi     = (const float*)d_in[1];
  const int*   labels = (const int*)d_in[2];
  float*       out    = (float*)d_out;

  float* part = (float*)d_ws;                          // [2][16][36864]
  float* lum  = part + (size_t)2 * CSPLIT * LUME;      // [2][36864]

  lum_partial_kernel<<<dim3((NB * P4) / 256, CSPLIT, 2), 256, 0, stream>>>(fv, fi, part);
  lum_reduce_kernel<<<(2 * LUME + 255) / 256, 256, 0, stream>>>(part, lum);
  pair_loss_kernel<<<1, 512, 0, stream>>>(lum, labels, out);
}